// BatchedRadiusGraphBuilder_56564719289125
// MI455X (gfx1250) — compile-verified
//
#include <hip/hip_runtime.h>
#include <stdint.h>

// Problem constants (from reference setup_inputs)
#define BATCH    16
#define NPT      1024
#define ROWS     (BATCH * NPT)     // 16384
#define STRIPS   (ROWS / 16)       // 1024 (one wave per 16-row strip)
#define TILES    (NPT / 16)        // 64 column tiles per strip
#define MAXE     1000000
#define CUT2     0.25f             // (0.5)^2
#define EPS2     1e-16f            // (1e-8)^2

typedef __attribute__((ext_vector_type(2))) float v2f;
typedef __attribute__((ext_vector_type(8))) float v8f;

// ---------------------------------------------------------------------------
// Kernel 0: zero the whole output region (padding entries must be 0).
// 5e6 dwords = 20MB -> HBM-bandwidth bound, wide b128 stores.
// ---------------------------------------------------------------------------
__global__ void fill_zero_kernel(float4* __restrict__ out, int n4) {
    int i = blockIdx.x * blockDim.x + threadIdx.x;
    int stride = gridDim.x * blockDim.x;
    float4 z = make_float4(0.f, 0.f, 0.f, 0.f);
    for (; i < n4; i += stride) out[i] = z;
}

// ---------------------------------------------------------------------------
// Kernel 1: per-row edge counts via V_WMMA_F32_16X16X4_F32 Gram tiles.
// Wave handles strip s = 16 rows of one batch; loops 64 column tiles.
//   A (16x4 f32): lanes 0-15 hold (x,y) of row m=lane%16, lanes 16-31 hold (z,0)
//   B (4x16 f32): same pattern for columns             -> C = P * P^T (16x16)
//   d^2 = |p_m|^2 + |p_n|^2 - 2*C[m][n]
// C layout: VGPR v, lane half h -> row v+8h, col lane%16. Ballot+popc per row.
// ---------------------------------------------------------------------------
__global__ __launch_bounds__(256)
void count_kernel(const float* __restrict__ pos,
                  const unsigned char* __restrict__ mask,
                  unsigned* __restrict__ counts) {
    const int lane  = threadIdx.x & 31;
    const int wave  = threadIdx.x >> 5;
    const int strip = blockIdx.x * 8 + wave;        // 0..1023
    const int b     = strip >> 6;                   // strip / (NPT/16)
    const int rowBase = (strip & 63) << 4;          // node index base in batch
    const int m     = lane & 15;
    const int half  = lane >> 4;

    const float* pb = pos + (size_t)b * NPT * 3;
    const unsigned char* mb = mask + (size_t)b * NPT;

    // Row point for this lane (row = rowBase + m), duplicated across halves.
    const int ar = rowBase + m;
    float ax = pb[ar * 3 + 0], ay = pb[ar * 3 + 1], az = pb[ar * 3 + 2];
    float ra = ax * ax + ay * ay + az * az;
    int   am = mb[ar] ? 1 : 0;

    v2f A;
    A.x = half ? az : ax;
    A.y = half ? 0.f : ay;

    // Broadcast norms/masks of the row this lane accumulates (row v + 8*half).
    float    rn[8];
    int      mrow[8];
#pragma unroll
    for (int v = 0; v < 8; ++v) {
        int src = v + 8 * half;
        rn[v]   = __shfl(ra, src, 32);
        mrow[v] = __shfl(am, src, 32);
    }

    unsigned cnt[8];
#pragma unroll
    for (int v = 0; v < 8; ++v) cnt[v] = 0u;

    for (int t = 0; t < TILES; ++t) {
        const int col = t * 16 + m;
        float bx = pb[col * 3 + 0], by = pb[col * 3 + 1], bz = pb[col * 3 + 2];
        float rc = bx * bx + by * by + bz * bz;
        int   mc = mb[col] ? 1 : 0;

        v2f Bm;
        Bm.x = half ? bz : bx;
        Bm.y = half ? 0.f : by;

        v8f C = {};
        C = __builtin_amdgcn_wmma_f32_16x16x4_f32(
                false, A, false, Bm, (short)0, C, false, false);

#pragma unroll
        for (int v = 0; v < 8; ++v) {
            float d2 = rn[v] + rc - 2.0f * C[v];
            bool pred = (d2 <= CUT2) && (d2 > EPS2) && mrow[v] && mc;
            unsigned bits = (unsigned)__ballot(pred);
            cnt[v] += half ? __popc(bits >> 16) : __popc(bits & 0xFFFFu);
        }
    }

    // Lanes 0 and 16 publish the 8 row counts of their half.
    if (m == 0) {
#pragma unroll
        for (int v = 0; v < 8; ++v)
            counts[strip * 16 + v + 8 * half] = cnt[v];
    }
}

// ---------------------------------------------------------------------------
// Kernel 2: exclusive prefix sum over 16384 row counts (single 1024-thread
// block, 16 rows/thread, Hillis-Steele block scan in LDS).
// ---------------------------------------------------------------------------
__global__ __launch_bounds__(1024)
void scan_kernel(const unsigned* __restrict__ counts,
                 unsigned* __restrict__ offsets) {
    __shared__ unsigned sums[1024];
    const int t = threadIdx.x;

    unsigned local[16];
    unsigned s = 0;
#pragma unroll
    for (int i = 0; i < 16; ++i) { local[i] = counts[t * 16 + i]; s += local[i]; }
    sums[t] = s;
    __syncthreads();

    for (int off = 1; off < 1024; off <<= 1) {
        unsigned v = (t >= off) ? sums[t - off] : 0u;
        __syncthreads();
        sums[t] += v;
        __syncthreads();
    }

    unsigned excl = (t == 0) ? 0u : sums[t - 1];
#pragma unroll
    for (int i = 0; i < 16; ++i) { offsets[t * 16 + i] = excl; excl += local[i]; }
}

// ---------------------------------------------------------------------------
// Kernel 3: ordered scatter. Identical WMMA predicate as count_kernel so the
// scan offsets are exact. Within a tile, a lane's write slot is
// rowOffset + runningCount + popc(row-ballot bits below this lane) which
// reproduces jnp.where's lexicographic (b, src, dst) order exactly.
// ---------------------------------------------------------------------------
__global__ __launch_bounds__(256)
void scatter_kernel(const float* __restrict__ pos,
                    const unsigned char* __restrict__ mask,
                    const unsigned* __restrict__ offsets,
                    int* __restrict__ esrc,
                    int* __restrict__ edst,
                    float* __restrict__ evec) {
    const int lane  = threadIdx.x & 31;
    const int wave  = threadIdx.x >> 5;
    const int strip = blockIdx.x * 8 + wave;
    const int b     = strip >> 6;
    const int rowBase = (strip & 63) << 4;
    const int m     = lane & 15;
    const int half  = lane >> 4;

    const float* pb = pos + (size_t)b * NPT * 3;
    const unsigned char* mb = mask + (size_t)b * NPT;

    const int ar = rowBase + m;
    float ax = pb[ar * 3 + 0], ay = pb[ar * 3 + 1], az = pb[ar * 3 + 2];
    float ra = ax * ax + ay * ay + az * az;
    int   am = mb[ar] ? 1 : 0;

    v2f A;
    A.x = half ? az : ax;
    A.y = half ? 0.f : ay;

    float    rn[8], rx[8], ry[8], rz[8];
    int      mrow[8];
    unsigned offv[8], run[8];
#pragma unroll
    for (int v = 0; v < 8; ++v) {
        int src = v + 8 * half;
        rn[v]   = __shfl(ra, src, 32);
        rx[v]   = __shfl(ax, src, 32);
        ry[v]   = __shfl(ay, src, 32);
        rz[v]   = __shfl(az, src, 32);
        mrow[v] = __shfl(am, src, 32);
        offv[v] = offsets[strip * 16 + src];
        run[v]  = 0u;
    }

    for (int t = 0; t < TILES; ++t) {
        const int col = t * 16 + m;
        float bx = pb[col * 3 + 0], by = pb[col * 3 + 1], bz = pb[col * 3 + 2];
        float rc = bx * bx + by * by + bz * bz;
        int   mc = mb[col] ? 1 : 0;

        v2f Bm;
        Bm.x = half ? bz : bx;
        Bm.y = half ? 0.f : by;

        v8f C = {};
        C = __builtin_amdgcn_wmma_f32_16x16x4_f32(
                false, A, false, Bm, (short)0, C, false, false);

#pragma unroll
        for (int v = 0; v < 8; ++v) {
            float d2 = rn[v] + rc - 2.0f * C[v];
            bool pred = (d2 <= CUT2) && (d2 > EPS2) && mrow[v] && mc;
            unsigned bits    = (unsigned)__ballot(pred);
            unsigned rowbits = half ? (bits >> 16) : (bits & 0xFFFFu);
            unsigned prefix  = __popc(rowbits & ((1u << m) - 1u));

            if (pred) {
                unsigned idx = offv[v] + run[v] + prefix;
                if (idx < (unsigned)MAXE) {
                    esrc[idx] = b * NPT + rowBase + v + 8 * half;
                    edst[idx] = b * NPT + col;
                    evec[3 * idx + 0] = bx - rx[v];
                    evec[3 * idx + 1] = by - ry[v];
                    evec[3 * idx + 2] = bz - rz[v];
                }
            }
            run[v] += __popc(rowbits);
        }
    }
}

// ---------------------------------------------------------------------------
extern "C" void kernel_launch(void* const* d_in, const int* in_sizes, int n_in,
                              void* d_out, int out_size, void* d_ws, size_t ws_size,
                              hipStream_t stream) {
    const float*         pos  = (const float*)d_in[0];
    const unsigned char* mask = (const unsigned char*)d_in[1];

    // Output layout: edge_src[1M] i32 | edge_dst[1M] i32 | edge_vec[1M*3] f32
    int*   esrc = (int*)d_out;
    int*   edst = esrc + MAXE;
    float* evec = (float*)(edst + MAXE);

    // Workspace: row counts + row offsets (2 * 16384 u32 = 128 KB)
    unsigned* counts  = (unsigned*)d_ws;
    unsigned* offsets = counts + ROWS;

    // 1) zero entire 20MB output (padding entries are zeros)
    const int n4 = (5 * MAXE) / 4;   // 5e6 dwords -> 1.25e6 float4
    fill_zero_kernel<<<2048, 256, 0, stream>>>((float4*)d_out, n4);

    // 2) WMMA Gram-matrix per-row counts (1024 strips, 8 waves/block)
    count_kernel<<<STRIPS / 8, 256, 0, stream>>>(pos, mask, counts);

    // 3) exclusive scan over 16384 rows (single block)
    scan_kernel<<<1, 1024, 0, stream>>>(counts, offsets);

    // 4) ordered WMMA scatter
    scatter_kernel<<<STRIPS / 8, 256, 0, stream>>>(pos, mask, offsets,
                                                   esrc, edst, evec);
}